// CrossModalAttn_20486994002055
// MI455X (gfx1250) — compile-verified
//
#include <hip/hip_runtime.h>
#include <hip/hip_bf16.h>

typedef __attribute__((ext_vector_type(2))) float v2f;
typedef __attribute__((ext_vector_type(4))) float v4f;
typedef __attribute__((ext_vector_type(8))) float v8f;

#define CHI 32
#define CHY 16
#define HH  256
#define WW  256
#define HW  (HH * WW)
#define CST 260   // c_lds row stride (pad to dodge bank conflicts, keeps 16B align)
#define SST 257   // s_buf row stride

static __device__ __forceinline__ v8f wmma_f32_4(v2f a, v2f b, v8f c) {
    return __builtin_amdgcn_wmma_f32_16x16x4_f32(
        /*neg_a=*/false, a, /*neg_b=*/false, b,
        /*c_mod=*/(short)0, c, /*reuse_a=*/false, /*reuse_b=*/false);
}

__global__ __launch_bounds__(256) void CrossModalAttn_kernel(
    const float* __restrict__ c,  const float* __restrict__ p,
    const float* __restrict__ Wq, const float* __restrict__ bq,
    const float* __restrict__ Wk, const float* __restrict__ bk,
    const float* __restrict__ Wv, const float* __restrict__ bv,
    const float* __restrict__ Wy, const float* __restrict__ by,
    float* __restrict__ out)
{
    __shared__ float c_lds[CHI * CST];   // staged c row: [32 ch][256 px]
    __shared__ float s_buf[CHY * SST];   // q*k scores, later reused for y16
    __shared__ float red[256];
    __shared__ float maxv[CHY];
    __shared__ float invs[CHY];

    const int t    = threadIdx.x;
    const int b    = blockIdx.x >> 8;
    const int h    = blockIdx.x & 255;
    const int lane = t & 31;
    const int wid  = t >> 5;            // wave id 0..7
    const int half = (lane >> 4) & 1;   // half-wave select
    const int col  = lane & 15;         // N / pixel-in-tile, also A-row M

    const size_t cBase = (size_t)b * CHI * HW + (size_t)h * WW;   // c[b][0][h][0]
    const size_t pBase = cBase;                                    // same geometry
    const size_t oBase = (size_t)b * (2 * CHI) * HW + (size_t)h * WW;

    // ---- 1. stage c row into LDS (coalesced float4) --------------------
    #pragma unroll
    for (int it = 0; it < 8; ++it) {
        int idx = t + it * 256;            // 0..2047 float4 slots
        int ch  = idx >> 6;                // 0..31
        int px4 = (idx & 63) << 2;         // 0..252
        v4f val = *reinterpret_cast<const v4f*>(c + cBase + (size_t)ch * HW + px4);
        *reinterpret_cast<v4f*>(&c_lds[ch * CST + px4]) = val;
    }

    // ---- weights/bias into registers (documented A / C layouts) --------
    v2f aq[8], ak[8], av[8];
    #pragma unroll
    for (int kb = 0; kb < 8; ++kb) {
        int ch0 = kb * 4 + half * 2;       // A: lane M=col, K slice per half
        aq[kb] = v2f{Wq[col * CHI + ch0], Wq[col * CHI + ch0 + 1]};
        ak[kb] = v2f{Wk[col * CHI + ch0], Wk[col * CHI + ch0 + 1]};
        av[kb] = v2f{Wv[col * CHI + ch0], Wv[col * CHI + ch0 + 1]};
    }
    v2f ay0[4], ay1[4];
    #pragma unroll
    for (int kb = 0; kb < 4; ++kb) {
        int ch0 = kb * 4 + half * 2;
        ay0[kb] = v2f{Wy[col * CHY + ch0],        Wy[col * CHY + ch0 + 1]};
        ay1[kb] = v2f{Wy[(col + 16) * CHY + ch0], Wy[(col + 16) * CHY + ch0 + 1]};
    }
    float bqv[8], bkv[8], bvv[8], by0[8], by1[8];
    #pragma unroll
    for (int r = 0; r < 8; ++r) {          // C/D layout: M = r + 8*half
        int ch = r + half * 8;
        bqv[r] = bq[ch]; bkv[r] = bk[ch]; bvv[r] = bv[ch];
        by0[r] = by[ch]; by1[r] = by[16 + ch];
    }
    __syncthreads();

    // ---- 2. q/k/v convs via chained fp32 WMMA, s = q*k -----------------
    v8f s_reg[2], v_reg[2];
    #pragma unroll
    for (int ti = 0; ti < 2; ++ti) {
        const int tile = wid * 2 + ti;
        const int px   = tile * 16 + col;
        v8f qa, ka, va;
        #pragma unroll
        for (int r = 0; r < 8; ++r) { qa[r] = bqv[r]; ka[r] = bkv[r]; va[r] = bvv[r]; }
        #pragma unroll
        for (int kb = 0; kb < 8; ++kb) {
            int ch0 = kb * 4 + half * 2;
            v2f bc, bp;
            bc.x = c_lds[ch0 * CST + px];
            bc.y = c_lds[(ch0 + 1) * CST + px];
            bp.x = p[pBase + (size_t)ch0 * HW + px];
            bp.y = p[pBase + (size_t)(ch0 + 1) * HW + px];
            qa = wmma_f32_4(aq[kb], bc, qa);
            ka = wmma_f32_4(ak[kb], bp, ka);
            va = wmma_f32_4(av[kb], bp, va);
        }
        v8f s;
        #pragma unroll
        for (int r = 0; r < 8; ++r) s[r] = qa[r] * ka[r];
        s_reg[ti] = s; v_reg[ti] = va;
        #pragma unroll
        for (int r = 0; r < 8; ++r)
            s_buf[(r + half * 8) * SST + px] = s[r];
    }
    __syncthreads();

    // ---- 3. softmax stats over W (per channel row of 256) --------------
    {
        int ch = t & 15, seg = t >> 4;
        float m0 = -__builtin_inff();
        #pragma unroll
        for (int i = 0; i < 16; ++i)
            m0 = fmaxf(m0, s_buf[ch * SST + seg * 16 + i]);
        red[seg * 16 + ch] = m0;
    }
    __syncthreads();
    if (t < 16) {
        float m0 = -__builtin_inff();
        #pragma unroll
        for (int seg = 0; seg < 16; ++seg) m0 = fmaxf(m0, red[seg * 16 + t]);
        maxv[t] = m0;
    }
    __syncthreads();
    {
        int ch = t & 15, seg = t >> 4;
        float mx = maxv[ch], s0 = 0.f;
        #pragma unroll
        for (int i = 0; i < 16; ++i)
            s0 += __expf(s_buf[ch * SST + seg * 16 + i] - mx);
        red[seg * 16 + ch] = s0;
    }
    __syncthreads();
    if (t < 16) {
        float s0 = 0.f;
        #pragma unroll
        for (int seg = 0; seg < 16; ++seg) s0 += red[seg * 16 + t];
        invs[t] = 1.0f / s0;
    }
    __syncthreads();

    // ---- 4. y16 = softmax(s) * v, written in B-operand order -----------
    #pragma unroll
    for (int ti = 0; ti < 2; ++ti) {
        const int px = (wid * 2 + ti) * 16 + col;
        v8f s = s_reg[ti], vv = v_reg[ti];
        #pragma unroll
        for (int r = 0; r < 8; ++r) {
            int ch = r + half * 8;
            s_buf[ch * SST + px] = __expf(s[r] - maxv[ch]) * invs[ch] * vv[r];
        }
    }
    __syncthreads();

    // ---- 5. final 32x16 conv via WMMA, store y then copy c -------------
    #pragma unroll
    for (int ti = 0; ti < 2; ++ti) {
        const int px = (wid * 2 + ti) * 16 + col;
        v8f y0, y1;
        #pragma unroll
        for (int r = 0; r < 8; ++r) { y0[r] = by0[r]; y1[r] = by1[r]; }
        #pragma unroll
        for (int kb = 0; kb < 4; ++kb) {
            int ch0 = kb * 4 + half * 2;
            v2f bb;
            bb.x = s_buf[ch0 * SST + px];
            bb.y = s_buf[(ch0 + 1) * SST + px];
            y0 = wmma_f32_4(ay0[kb], bb, y0);
            y1 = wmma_f32_4(ay1[kb], bb, y1);
        }
        #pragma unroll
        for (int r = 0; r < 8; ++r) {
            int ch = r + half * 8;
            out[oBase + (size_t)ch * HW + px]        = y0[r];
            out[oBase + (size_t)(ch + 16) * HW + px] = y1[r];
        }
    }
    #pragma unroll
    for (int it = 0; it < 8; ++it) {
        int idx = t + it * 256;
        int ch  = idx >> 6;
        int px4 = (idx & 63) << 2;
        v4f val = *reinterpret_cast<const v4f*>(&c_lds[ch * CST + px4]);
        *reinterpret_cast<v4f*>(out + oBase + (size_t)(CHI + ch) * HW + px4) = val;
    }
}

extern "C" void kernel_launch(void* const* d_in, const int* in_sizes, int n_in,
                              void* d_out, int out_size, void* d_ws, size_t ws_size,
                              hipStream_t stream) {
    (void)in_sizes; (void)n_in; (void)out_size; (void)d_ws; (void)ws_size;
    const float* c  = (const float*)d_in[0];
    const float* p  = (const float*)d_in[1];
    const float* Wq = (const float*)d_in[2];
    const float* bq = (const float*)d_in[3];
    const float* Wk = (const float*)d_in[4];
    const float* bk = (const float*)d_in[5];
    const float* Wv = (const float*)d_in[6];
    const float* bv = (const float*)d_in[7];
    const float* Wy = (const float*)d_in[8];
    const float* by = (const float*)d_in[9];
    float* out = (float*)d_out;

    dim3 grid(16 * 256);   // one block per (b, h) row
    dim3 block(256);       // 8 wave32 waves
    CrossModalAttn_kernel<<<grid, block, 0, stream>>>(
        c, p, Wq, bq, Wk, bk, Wv, bv, Wy, by, out);
}